// CustomCrossEntropyLoss_89756226552392
// MI455X (gfx1250) — compile-verified
//
#include <hip/hip_runtime.h>
#include <hip/hip_bf16.h>
#include <cstdint>

#define BATCH 8192
#define NCLS  32000
#define NV4   (NCLS / 4)   // 8000 float4 chunks per row
#define TPB   1024
#define NWAVE (TPB / 32)   // 32 waves (wave32)

__device__ __forceinline__ float wave_max32(float v) {
#pragma unroll
  for (int off = 16; off >= 1; off >>= 1)
    v = fmaxf(v, __shfl_xor(v, off, 32));
  return v;
}

__device__ __forceinline__ float wave_sum32(float v) {
#pragma unroll
  for (int off = 16; off >= 1; off >>= 1)
    v += __shfl_xor(v, off, 32);
  return v;
}

// One workgroup per row: async-stage the 128 KB row into LDS once,
// then compute logsumexp entirely from LDS.
__global__ __launch_bounds__(TPB) void ce_row_kernel(
    const float* __restrict__ x, const int* __restrict__ y0,
    const float* __restrict__ a1, const int* __restrict__ gramma,
    float* __restrict__ row_loss) {
  __shared__ float row[NCLS];     // 128000 B (WGP LDS is 320 KB)
  __shared__ float red[NWAVE];

  const int b   = blockIdx.x;
  const int tid = threadIdx.x;
  const float* src = x + (size_t)b * NCLS;   // uniform -> SGPR pair

  // ---- CDNA5 async HBM -> LDS copy (ASYNCcnt path) ----
  // GVS mode: mem_addr = SADDR(64b sgpr) + VADDR(32b byte offset)
  // VDST vgpr = per-lane LDS byte address (low 32 bits of flat LDS addr).
  for (int i = tid; i < NV4; i += TPB) {
    uint32_t lds_addr = (uint32_t)(uintptr_t)(&row[i * 4]);
    uint32_t goff     = (uint32_t)(i * 16);
    asm volatile("global_load_async_to_lds_b128 %0, %1, %2"
                 :: "v"(lds_addr), "v"(goff), "s"(src)
                 : "memory");
  }
  asm volatile("s_wait_asynccnt 0" ::: "memory");
  __syncthreads();

  const float4* row4 = (const float4*)row;

  // ---- pass 1: row max (ds_load_b128 vectorized) ----
  float m = -3.402823466e38f;
  for (int i = tid; i < NV4; i += TPB) {
    float4 v = row4[i];
    m = fmaxf(m, fmaxf(fmaxf(v.x, v.y), fmaxf(v.z, v.w)));
  }
  m = wave_max32(m);
  const int wave = tid >> 5, lane = tid & 31;
  if (lane == 0) red[wave] = m;
  __syncthreads();
  if (wave == 0) {
    float t = red[lane];          // NWAVE == 32 exactly
    t = wave_max32(t);
    if (lane == 0) red[0] = t;
  }
  __syncthreads();
  m = red[0];
  __syncthreads();                // red is reused below

  // ---- pass 2: sum of exp(x - m) ----
  float s = 0.f;
  for (int i = tid; i < NV4; i += TPB) {
    float4 v = row4[i];
    s += __expf(v.x - m) + __expf(v.y - m) + __expf(v.z - m) + __expf(v.w - m);
  }
  s = wave_sum32(s);
  if (lane == 0) red[wave] = s;
  __syncthreads();

  if (tid == 0) {
    float tot = 0.f;
#pragma unroll
    for (int w = 0; w < NWAVE; ++w) tot += red[w];
    const int y = y0[b];
    const float logp = row[y] - m - __logf(tot);      // log_softmax picked
    const float wt   = powf(2.0f * a1[b], (float)gramma[0]);
    row_loss[b] = wt * logp;
  }
}

// Deterministic tree reduction of the 8192 per-row terms -> scalar.
__global__ __launch_bounds__(1024) void ce_finalize(
    const float* __restrict__ row_loss, float* __restrict__ out) {
  __shared__ float red[32];
  float s = 0.f;
  for (int i = threadIdx.x; i < BATCH; i += 1024) s += row_loss[i];
  s = wave_sum32(s);
  const int wave = threadIdx.x >> 5, lane = threadIdx.x & 31;
  if (lane == 0) red[wave] = s;
  __syncthreads();
  if (threadIdx.x == 0) {
    float tot = 0.f;
#pragma unroll
    for (int w = 0; w < 32; ++w) tot += red[w];
    out[0] = -tot / (float)BATCH;
  }
}

extern "C" void kernel_launch(void* const* d_in, const int* in_sizes, int n_in,
                              void* d_out, int out_size, void* d_ws, size_t ws_size,
                              hipStream_t stream) {
  const float* x      = (const float*)d_in[0];
  const int*   y0     = (const int*)d_in[1];
  const float* a1     = (const float*)d_in[2];
  const int*   gramma = (const int*)d_in[3];
  float* row_loss = (float*)d_ws;   // 8192 floats, fully overwritten each call

  ce_row_kernel<<<BATCH, TPB, 0, stream>>>(x, y0, a1, gramma, row_loss);
  ce_finalize<<<1, 1024, 0, stream>>>(row_loss, (float*)d_out);
}